// Qwen3MoeAttentionLayer_21861383537427
// MI455X (gfx1250) — compile-verified
//
#include <hip/hip_runtime.h>
#include <hip/hip_bf16.h>

typedef __attribute__((ext_vector_type(2))) float v2f;
typedef __attribute__((ext_vector_type(8))) float v8f;

#define NEG_INF (-__builtin_inff())

namespace cfg {
constexpr int B = 32, S = 4096, H = 2048;
constexpr int HQ = 32, HKV = 4, D = 128, G = HQ / HKV;
constexpr int PAGE = 32, PPS = S / PAGE;       // pages per sequence = 128
constexpr int CHUNK = 32;                       // keys per flash chunk
constexpr int SPLIT = 4;                        // sequence splits (flash-decoding)
constexpr int SLEN = S / SPLIT;                 // 1024 keys per split
constexpr float EPS = 1e-6f;
constexpr float LOG_THETA = 16.11809565095832f; // ln(1e7)
}

// ---------------------------------------------------------------------------
// Async global -> LDS copy (16B per lane), tracked by ASYNCcnt.
// ---------------------------------------------------------------------------
__device__ __forceinline__ void async_copy_b128(void* lds_dst, const void* src) {
    unsigned off = (unsigned)(uintptr_t)lds_dst;   // low 32 bits = LDS offset
    asm volatile("global_load_async_to_lds_b128 %0, %1, off"
                 :: "v"(off), "v"(src) : "memory");
}
__device__ __forceinline__ void wait_async_le16() {
    asm volatile("s_wait_asynccnt 16" ::: "memory");
}
__device__ __forceinline__ void wait_async_0() {
    asm volatile("s_wait_asynccnt 0" ::: "memory");
}

// ---------------------------------------------------------------------------
// C[32 x N] = A[32 x K] @ W[K x N], fp32. One wave per 16-wide N tile; both
// 16-row M tiles computed in the same wave so each W fragment feeds 2 WMMAs.
// grid = (N/16), block = 32 (one wave -> EXEC all ones for WMMA).
// ---------------------------------------------------------------------------
__global__ void gemm_wmma(const float* __restrict__ A, const float* __restrict__ W,
                          float* __restrict__ C, int K, int N) {
    const int lane = threadIdx.x;          // 0..31
    const int l16  = lane & 15;
    const int half = lane >> 4;            // 0 or 1
    const int nBase = blockIdx.x * 16;
    const int n = nBase + l16;

    v8f c0 = {}, c1 = {};
    for (int k = 0; k < K; k += 4) {
        const int ka = k + 2 * half;       // lane<16 -> K{0,1}; lane>=16 -> K{2,3}
        v2f a0 = *(const v2f*)(A + (size_t)l16 * K + ka);
        v2f a1 = *(const v2f*)(A + (size_t)(16 + l16) * K + ka);
        v2f b;
        b.x = W[(size_t)(ka + 0) * N + n];
        b.y = W[(size_t)(ka + 1) * N + n];
        c0 = __builtin_amdgcn_wmma_f32_16x16x4_f32(false, a0, false, b,
                                                   (short)0, c0, false, false);
        c1 = __builtin_amdgcn_wmma_f32_16x16x4_f32(false, a1, false, b,
                                                   (short)0, c1, false, false);
    }
#pragma unroll
    for (int r = 0; r < 8; ++r) {
        const int mm = r + 8 * half;       // C layout: row r (lanes<16) / r+8
        C[(size_t)mm * N + n]        = c0[r];
        C[(size_t)(16 + mm) * N + n] = c1[r];
    }
}

// ---------------------------------------------------------------------------
// Per-(b,head) RMSNorm + RoPE, in place. grid = B*(HQ+HKV), block = 128.
// ---------------------------------------------------------------------------
__global__ void norm_rope(float* __restrict__ qb, float* __restrict__ kb,
                          const float* __restrict__ qw, const float* __restrict__ kw,
                          const long long* __restrict__ positions) {
    using namespace cfg;
    const int bid = blockIdx.x;
    const int b = bid / (HQ + HKV);
    const int h = bid % (HQ + HKV);
    float* x;
    const float* w;
    if (h < HQ) { x = qb + ((size_t)b * HQ + h) * D;        w = qw; }
    else        { x = kb + ((size_t)b * HKV + (h - HQ)) * D; w = kw; }

    const int d = threadIdx.x;            // 0..127
    float v = x[d];
    float s = v * v;
#pragma unroll
    for (int off = 16; off > 0; off >>= 1) s += __shfl_xor(s, off);

    __shared__ float partial[4];
    __shared__ float sh[D];
    if ((threadIdx.x & 31) == 0) partial[threadIdx.x >> 5] = s;
    __syncthreads();
    const float tot = partial[0] + partial[1] + partial[2] + partial[3];
    const float rstd = rsqrtf(tot / (float)D + EPS);
    const float xn = v * rstd * w[d];
    sh[d] = xn;
    __syncthreads();

    const float pos = (float)positions[b];
    const int i = d & 63;
    const float inv_freq = __expf(-((float)(2 * i) / (float)D) * LOG_THETA);
    const float ang = pos * inv_freq;
    const float cs = __cosf(ang), sn = __sinf(ang);
    float out;
    if (d < 64) out = xn * cs - sh[d + 64] * sn;
    else        out = xn * cs + sh[d - 64] * sn;
    x[d] = out;
}

// ---------------------------------------------------------------------------
// Flash-decode attention, split over the sequence. grid = B*HKV*SPLIT,
// block = 128. Each block covers SLEN keys and emits an UNNORMALIZED partial
// (o_part, m_part, l_part). K/V chunks are double-buffered in LDS via
// global_load_async_to_lds_b128; scores and P@V via V_WMMA_F32_16X16X4_F32.
// ---------------------------------------------------------------------------
__global__ void attn_decode(const float* __restrict__ qb,
                            const float* __restrict__ kn, const float* __restrict__ vn,
                            const float* __restrict__ kc, const float* __restrict__ vc,
                            const long long* __restrict__ positions,
                            const int* __restrict__ page_table,
                            float* __restrict__ o_part,
                            float* __restrict__ m_part,
                            float* __restrict__ l_part) {
    using namespace cfg;
    const int sp  = blockIdx.x % SPLIT;
    const int bk  = blockIdx.x / SPLIT;        // b*HKV + kvh
    const int b   = bk / HKV;
    const int kvh = bk % HKV;
    const int tid  = threadIdx.x;       // 0..127
    const int lane = tid & 31;
    const int wave = tid >> 5;          // 0..3
    const int l16  = lane & 15;
    const int half = lane >> 4;
    const long long pos = positions[b];
    const float scale = rsqrtf((float)D);
    const int sBeg = sp * SLEN;

    __shared__ float Qs[G * D];            // 8 x 128      (4 KB)
    __shared__ float Ks[2][CHUNK * D];     // 2 x 32 x 128 (32 KB)
    __shared__ float Vs[2][CHUNK * D];     //              (32 KB)
    __shared__ float Ss[G * CHUNK];
    __shared__ float Ps[G * CHUNK];
    __shared__ float Ms[G], Ls[G], Al[G];

    for (int i = tid; i < G * D; i += 128) {
        int g = i / D, d = i % D;
        Qs[i] = qb[(((size_t)b * HQ) + kvh * G + g) * D + d];
    }
    if (tid < G) { Ms[tid] = NEG_INF; Ls[tid] = 0.f; }

    // issues 16 async ops per thread (8 x {K,V})
    auto prefetch = [&](float* Kdst, float* Vdst, int s0c) {
        for (int i = tid; i < CHUNK * (D / 4); i += 128) {
            const int key = i / (D / 4);
            const int d4  = (i % (D / 4)) * 4;
            const long long sg = (long long)(s0c + key);
            const float *ksrc, *vsrc;
            if (sg == pos) {
                const size_t src = (((size_t)b * HKV) + kvh) * D + d4;
                ksrc = kn + src; vsrc = vn + src;
            } else {
                const int page = page_table[b * PPS + (int)(sg >> 5)];
                const size_t src =
                    ((((size_t)page * PAGE + (sg & 31)) * HKV) + kvh) * D + d4;
                ksrc = kc + src; vsrc = vc + src;
            }
            async_copy_b128(Kdst + key * D + d4, ksrc);
            async_copy_b128(Vdst + key * D + d4, vsrc);
        }
    };

    prefetch(Ks[0], Vs[0], sBeg);

    v8f co0 = {}, co1 = {};
    const int col0 = wave * 32;         // this wave owns D columns [col0, col0+32)

    for (int s0 = sBeg; s0 < sBeg + SLEN; s0 += CHUNK) {
        const int  cur  = ((s0 - sBeg) / CHUNK) & 1;
        const bool more = (s0 + CHUNK) < (sBeg + SLEN);
        if (more) prefetch(Ks[cur ^ 1], Vs[cur ^ 1], s0 + CHUNK);
        if (more) wait_async_le16();    // oldest 16 (current chunk) have landed
        else      wait_async_0();
        __syncthreads();
        const float* Kc = Ks[cur];
        const float* Vc = Vs[cur];

        // ---- scores: waves 0,1 each compute a 16-key tile of Q@K^T
        if (wave < 2) {
            const int keyBase = wave * 16;
            v8f cs = {};
            for (int k = 0; k < D; k += 4) {
                const int ka = k + 2 * half;
                const int mq = (l16 < G) ? l16 : 0;
                v2f a;
                a.x = Qs[mq * D + ka];
                a.y = Qs[mq * D + ka + 1];
                if (l16 >= G) { a.x = 0.f; a.y = 0.f; }
                const int n = keyBase + l16;
                v2f bb;
                bb.x = Kc[n * D + ka];
                bb.y = Kc[n * D + ka + 1];
                cs = __builtin_amdgcn_wmma_f32_16x16x4_f32(false, a, false, bb,
                                                           (short)0, cs, false, false);
            }
            if (half == 0) {
                const long long sg = (long long)(s0 + keyBase + l16);
                const float maskv = (sg <= pos) ? 0.f : NEG_INF;
#pragma unroll
                for (int r = 0; r < G; ++r)
                    Ss[r * CHUNK + keyBase + l16] = cs[r] * scale + maskv;
            }
        }
        __syncthreads();

        // ---- online softmax: each wave owns heads {wave, wave+4}
        for (int h = wave; h < G; h += 4) {
            const float val = Ss[h * CHUNK + lane];
            float mx = val;
#pragma unroll
            for (int off = 16; off > 0; off >>= 1) mx = fmaxf(mx, __shfl_xor(mx, off));
            const float mold = Ms[h];
            const float mnew = fmaxf(mold, mx);
            float p = (val == NEG_INF) ? 0.f : __expf(val - mnew);
            float ps = p;
#pragma unroll
            for (int off = 16; off > 0; off >>= 1) ps += __shfl_xor(ps, off);
            const float alpha = (mold == NEG_INF) ? 0.f : __expf(mold - mnew);
            Ps[h * CHUNK + lane] = p;
            if (lane == 0) { Ms[h] = mnew; Ls[h] = Ls[h] * alpha + ps; Al[h] = alpha; }
        }
        __syncthreads();

        // ---- P @ V via WMMA, rescale accumulator by alpha (pad rows stay 0)
#pragma unroll
        for (int r = 0; r < 8; ++r) {
            const float al = Al[r];
            co0[r] *= al;
            co1[r] *= al;
        }
        for (int kk = 0; kk < CHUNK; kk += 4) {
            const int ka = kk + 2 * half;
            const int mq = (l16 < G) ? l16 : 0;
            v2f a;
            a.x = Ps[mq * CHUNK + ka];
            a.y = Ps[mq * CHUNK + ka + 1];
            if (l16 >= G) { a.x = 0.f; a.y = 0.f; }
            const int n0 = col0 + l16;
            const int n1 = col0 + 16 + l16;
            v2f b0, b1;
            b0.x = Vc[ka * D + n0];       b0.y = Vc[(ka + 1) * D + n0];
            b1.x = Vc[ka * D + n1];       b1.y = Vc[(ka + 1) * D + n1];
            co0 = __builtin_amdgcn_wmma_f32_16x16x4_f32(false, a, false, b0,
                                                        (short)0, co0, false, false);
            co1 = __builtin_amdgcn_wmma_f32_16x16x4_f32(false, a, false, b1,
                                                        (short)0, co1, false, false);
        }
        __syncthreads();    // protect all LDS buffers for next chunk
    }

    // ---- epilogue: emit unnormalized partials + (m, l) per head
    const size_t pbase = ((size_t)bk * SPLIT + sp) * G;
    if (half == 0) {
#pragma unroll
        for (int r = 0; r < G; ++r) {
            const size_t ob = (pbase + r) * D;
            o_part[ob + col0 + l16]      = co0[r];
            o_part[ob + col0 + 16 + l16] = co1[r];
        }
    }
    if (tid < G) {
        m_part[pbase + tid] = Ms[tid];
        l_part[pbase + tid] = Ls[tid];
    }
}

// ---------------------------------------------------------------------------
// Combine SPLIT partials per (b, hq). grid = B*HQ, block = 128 (one per dim).
// ---------------------------------------------------------------------------
__global__ void attn_combine(const float* __restrict__ o_part,
                             const float* __restrict__ m_part,
                             const float* __restrict__ l_part,
                             float* __restrict__ ob) {
    using namespace cfg;
    const int blk = blockIdx.x;
    const int b   = blk / HQ;
    const int hq  = blk % HQ;
    const int kvh = hq / G;
    const int g   = hq % G;
    const int d   = threadIdx.x;

    const size_t base = (((size_t)b * HKV + kvh) * SPLIT) * G + g;
    float M = NEG_INF;
#pragma unroll
    for (int sp = 0; sp < SPLIT; ++sp)
        M = fmaxf(M, m_part[base + (size_t)sp * G]);

    float L = 0.f, acc = 0.f;
#pragma unroll
    for (int sp = 0; sp < SPLIT; ++sp) {
        const float l = l_part[base + (size_t)sp * G];
        const float m = m_part[base + (size_t)sp * G];
        const float w = (l == 0.f) ? 0.f : __expf(m - M);
        L   += l * w;
        acc += o_part[(base + (size_t)sp * G) * D + d] * w;
    }
    const float inv_l = (L > 0.f) ? (1.0f / L) : 0.f;
    ob[((size_t)b * HQ + hq) * D + d] = acc * inv_l;
}

// ---------------------------------------------------------------------------
extern "C" void kernel_launch(void* const* d_in, const int* in_sizes, int n_in,
                              void* d_out, int out_size, void* d_ws, size_t ws_size,
                              hipStream_t stream) {
    using namespace cfg;
    (void)in_sizes; (void)n_in; (void)out_size; (void)ws_size;

    const float*     hidden = (const float*)d_in[0];
    const float*     wq     = (const float*)d_in[1];
    const float*     wk     = (const float*)d_in[2];
    const float*     wv     = (const float*)d_in[3];
    const float*     wo     = (const float*)d_in[4];
    const float*     qnw    = (const float*)d_in[5];
    const float*     knw    = (const float*)d_in[6];
    const float*     kc     = (const float*)d_in[7];
    const float*     vc     = (const float*)d_in[8];
    const long long* pos    = (const long long*)d_in[9];
    const int*       pt     = (const int*)d_in[10];
    float*           out    = (float*)d_out;

    float* ws     = (float*)d_ws;
    float* q_buf  = ws;                              // B*HQ*D  = 131072
    float* k_buf  = q_buf + (size_t)B * HQ * D;      // B*HKV*D = 16384
    float* v_buf  = k_buf + (size_t)B * HKV * D;     // B*HKV*D = 16384
    float* o_buf  = v_buf + (size_t)B * HKV * D;     // B*HQ*D  = 131072
    float* o_part = o_buf + (size_t)B * HQ * D;      // B*HKV*SPLIT*G*D = 524288
    float* m_prt  = o_part + (size_t)B * HKV * SPLIT * G * D;   // 4096
    float* l_prt  = m_prt + (size_t)B * HKV * SPLIT * G;        // 4096

    // QKV projections
    gemm_wmma<<<HQ * D / 16, 32, 0, stream>>>(hidden, wq, q_buf, H, HQ * D);
    gemm_wmma<<<HKV * D / 16, 32, 0, stream>>>(hidden, wk, k_buf, H, HKV * D);
    gemm_wmma<<<HKV * D / 16, 32, 0, stream>>>(hidden, wv, v_buf, H, HKV * D);

    // RMSNorm + RoPE on q and k_new
    norm_rope<<<B * (HQ + HKV), 128, 0, stream>>>(q_buf, k_buf, qnw, knw, pos);

    // Paged flash-decode attention, split over sequence for HBM concurrency
    attn_decode<<<B * HKV * SPLIT, 128, 0, stream>>>(q_buf, k_buf, v_buf, kc, vc,
                                                     pos, pt, o_part, m_prt, l_prt);
    attn_combine<<<B * HQ, 128, 0, stream>>>(o_part, m_prt, l_prt, o_buf);

    // Output projection
    gemm_wmma<<<H / 16, 32, 0, stream>>>(o_buf, wo, out, HQ * D, H);
}